// ProtoNet_9672266350915
// MI455X (gfx1250) — compile-verified
//
#include <hip/hip_runtime.h>
#include <hip/hip_bf16.h>
#include <math.h>

// ---------------------------------------------------------------------------
// ProtoNet memory-augmented prototype + logits for MI455X (gfx1250, wave32).
// Dominant cost: sim_mem = l2norm(memory[300000x640]) . proto^T  -> HBM bound
// (768 MB @ 23.3 TB/s ~ 33us). Implemented with V_WMMA_F32_16X16X4_F32:
//   A = 16 memory rows x K=4 (fp32) streamed from HBM (global_load_b64),
//   B = K=4 x 16 cols (ways 0..4 used), resident in LDS (ds_load_b64,
//       conflict-free: 32 lanes x 2 dwords cover all 64 banks once).
// Row norms come for free from the A-fragment loads + one shfl_xor; a single
// reciprocal per row is shuffled to the D-layout lanes (no per-element div).
// ---------------------------------------------------------------------------

typedef float v2f __attribute__((ext_vector_type(2)));
typedef float v8f __attribute__((ext_vector_type(8)));

#define D_DIM   640
#define N_WAY   5
#define N_SHOT  5
#define N_QROWS 75          // 15 * 5
#define KTOP    16
#define KSTEPS  160         // 640 / 4
#define EPSF    1e-8f
#define NEG_INF (-3.0e38f)

// ---------------------------------------------------------------------------
// Kernel 1: basic proto (mean over shots, l2norm), sim_sup, and B-fragment
// pre-pack for the WMMA in kernel 2.
// B layout for V_WMMA_F32_16X16X4_F32 (mirrors documented A 16x4 layout):
//   lane L: N = L&15, VGPR0 holds K = (L<16 ? 0 : 2), VGPR1 holds K+1.
// bpack[kk*32 + L] = float2{ B[k][n], B[k+1][n] },  k = kk*4 + (L>>4)*2.
// ---------------------------------------------------------------------------
__global__ void pn_prep_kernel(const float* __restrict__ support,
                               float* __restrict__ ws_proto,   // [5*640]
                               float* __restrict__ ws_simsup,  // [5*5]  [w][s]
                               float* __restrict__ ws_bpack)   // [160*32*2]
{
    __shared__ float sProto[N_WAY * D_DIM];
    __shared__ float sSum[N_WAY];
    const int tid = threadIdx.x;

    // mean over shots (support layout: [shot][way][d])
    for (int idx = tid; idx < N_WAY * D_DIM; idx += blockDim.x) {
        const int w = idx / D_DIM, dd = idx - w * D_DIM;
        float acc = 0.f;
#pragma unroll
        for (int s = 0; s < N_SHOT; ++s)
            acc += support[(s * N_WAY + w) * D_DIM + dd];
        sProto[idx] = acc * (1.0f / N_SHOT);
    }
    if (tid < N_WAY) sSum[tid] = 0.f;
    __syncthreads();
    for (int idx = tid; idx < N_WAY * D_DIM; idx += blockDim.x) {
        const int w = idx / D_DIM;
        const float v = sProto[idx];
        atomicAdd(&sSum[w], v * v);
    }
    __syncthreads();
    for (int idx = tid; idx < N_WAY * D_DIM; idx += blockDim.x) {
        const int w = idx / D_DIM;
        const float nrm = fmaxf(sqrtf(sSum[w]), EPSF);
        const float p = sProto[idx] / nrm;
        sProto[idx] = p;
        ws_proto[idx] = p;
    }
    __syncthreads();

    // sim_sup[w][s] = dot(l2norm(support[s][w]), proto[w])
    if (tid < N_WAY * N_SHOT) {
        const int w = tid / N_SHOT, s = tid - w * N_SHOT;
        float dot = 0.f, ss = 0.f;
        const float* sp = support + (s * N_WAY + w) * D_DIM;
        for (int dd = 0; dd < D_DIM; ++dd) {
            const float v = sp[dd];
            dot += v * sProto[w * D_DIM + dd];
            ss += v * v;
        }
        ws_simsup[tid] = dot / fmaxf(sqrtf(ss), EPSF);
    }

    // B fragments for all 160 K-steps
    for (int i = tid; i < KSTEPS * 32; i += blockDim.x) {
        const int kk = i >> 5, L = i & 31;
        const int n = L & 15;
        const int k = kk * 4 + ((L >> 4) << 1);
        float b0 = 0.f, b1 = 0.f;
        if (n < N_WAY) {
            b0 = sProto[n * D_DIM + k];
            b1 = sProto[n * D_DIM + k + 1];
        }
        ws_bpack[2 * i] = b0;
        ws_bpack[2 * i + 1] = b1;
    }
}

// ---------------------------------------------------------------------------
// Kernel 2: HBM-bound WMMA stream. One wave per 16-row tile of memory.
// A frag (documented layout): lane L -> row M=L&15, K pair at kk*4+(L>>4)*2.
// D layout: lane L holds col N=L&15 of rows M = r + (L>=16 ? 8 : 0).
// simWS[w * Mpad + m] = 0.2 * dot(mem[m], proto[w]) / ||mem[m]||.
// B fragments live in LDS for the whole block (40 KB of the 320 KB WGP LDS),
// keeping the VMEM pipe + LOADcnt dedicated to the HBM A stream.
// ---------------------------------------------------------------------------
__global__ __launch_bounds__(256)
void pn_sim_kernel(const float* __restrict__ mem,
                   const float* __restrict__ bpack,
                   float* __restrict__ simWS,
                   int M, int Mpad)
{
    __shared__ float sB[KSTEPS * 32 * 2];   // 40 KB, block-invariant B frags

    // cooperative fill (coalesced b64 copies), once per block
    {
        const float2* src = (const float2*)bpack;
        float2* dst = (float2*)sB;
        for (int i = threadIdx.x; i < KSTEPS * 32; i += 256)
            dst[i] = src[i];
    }
    __syncthreads();

    const int lane   = threadIdx.x & 31;
    const int gwave  = (int)((blockIdx.x * blockDim.x + threadIdx.x) >> 5);
    const int nwaves = (int)((gridDim.x * blockDim.x) >> 5);
    const int tiles  = (M + 15) >> 4;

    const int mrow  = lane & 15;
    const int khalf = (lane >> 4) << 1;   // 0 or 2
    const float2* blds = (const float2*)sB + lane;

    for (int tile = gwave; tile < tiles; tile += nwaves) {
        const int m0 = tile << 4;
        const float* rowp = mem + (size_t)(m0 + mrow) * D_DIM + khalf;

        v8f acc = {0.f, 0.f, 0.f, 0.f, 0.f, 0.f, 0.f, 0.f};
        float sumsq = 0.f;

#pragma unroll 8
        for (int kk = 0; kk < KSTEPS; ++kk) {
            const float2 av = *(const float2*)(rowp + kk * 4);   // HBM stream
            const float2 bv = blds[kk * 32];                     // LDS, bank-clean
            v2f a; a.x = av.x; a.y = av.y;
            v2f b; b.x = bv.x; b.y = bv.y;
            sumsq = fmaf(av.x, av.x, sumsq);
            sumsq = fmaf(av.y, av.y, sumsq);
            acc = __builtin_amdgcn_wmma_f32_16x16x4_f32(
                false, a, false, b, (short)0, acc, false, false);
        }

        // full ||row||^2: this lane covers K%4 in {0,1} or {2,3}; partner lane
        // (lane ^ 16) covers the other half of the same row (L&15).
        const float other = __shfl_xor(sumsq, 16, 32);
        // ONE division per row, then broadcast the reciprocal to D-layout lanes
        const float rinv = 0.2f / fmaxf(sqrtf(sumsq + other), EPSF);

        const int hi = (lane >> 4) << 3;   // 0 or 8
        const int w  = lane & 15;
#pragma unroll
        for (int r = 0; r < 8; ++r) {
            const float ri = __shfl(rinv, r + hi, 32);   // 0.2/||row r+hi||
            const int m = m0 + r + hi;
            if (w < N_WAY && m < M)
                simWS[(size_t)w * Mpad + m] = acc[r] * ri;
        }
    }
}

// ---------------------------------------------------------------------------
// Kernel 3: per-way top-16 over [5 support sims ++ M memory sims].
// blockIdx.x = way. Thread-local sorted top-16 in registers (unrolled
// insertion network), then 16 rounds of block-wide argmax merge.
// ---------------------------------------------------------------------------
__global__ __launch_bounds__(256)
void pn_topk_kernel(const float* __restrict__ simWS,
                    const float* __restrict__ simsup,
                    float* __restrict__ topVal,   // [5*16]
                    int*   __restrict__ topIdx,   // [5*16]  (global concat idx)
                    int M, int Mpad)
{
    const int w = blockIdx.x;
    const int tid = threadIdx.x;

    float val[KTOP];
    int   idx[KTOP];
#pragma unroll
    for (int r = 0; r < KTOP; ++r) { val[r] = NEG_INF; idx[r] = -1; }

    const int total = M + N_SHOT;
    for (int i = tid; i < total; i += 256) {
        const float v = (i < N_SHOT) ? simsup[w * N_SHOT + i]
                                     : simWS[(size_t)w * Mpad + (i - N_SHOT)];
        if (v > val[KTOP - 1]) {
            float cv = v; int ci = i;
#pragma unroll
            for (int r = 0; r < KTOP; ++r) {
                const bool gt = cv > val[r];
                const float hiV = gt ? cv : val[r];
                const float loV = gt ? val[r] : cv;
                const int   hiI = gt ? ci : idx[r];
                const int   loI = gt ? idx[r] : ci;
                val[r] = hiV; idx[r] = hiI; cv = loV; ci = loI;
            }
        }
    }

    __shared__ float lv[256 * KTOP];
    __shared__ int   li[256 * KTOP];
    __shared__ float redV[256];
    __shared__ int   redT[256];
    __shared__ int   scur[256];

#pragma unroll
    for (int r = 0; r < KTOP; ++r) { lv[tid * KTOP + r] = val[r]; li[tid * KTOP + r] = idx[r]; }
    scur[tid] = 0;
    __syncthreads();

    for (int r = 0; r < KTOP; ++r) {
        const int c = scur[tid];
        redV[tid] = (c < KTOP) ? lv[tid * KTOP + c] : NEG_INF;
        redT[tid] = tid;
        __syncthreads();
        for (int s = 128; s > 0; s >>= 1) {
            if (tid < s && redV[tid + s] > redV[tid]) {
                redV[tid] = redV[tid + s];
                redT[tid] = redT[tid + s];
            }
            __syncthreads();
        }
        const int winner = redT[0];
        if (tid == 0) {
            topVal[w * KTOP + r] = redV[0];
            topIdx[w * KTOP + r] = li[winner * KTOP + scur[winner]];
        }
        __syncthreads();
        if (tid == winner) scur[tid]++;
        __syncthreads();
    }
}

// ---------------------------------------------------------------------------
// Kernel 4: gather top-16 embeddings, weighted prototype, l2norm, logits.
// ---------------------------------------------------------------------------
__global__ void pn_final_kernel(const float* __restrict__ support,
                                const float* __restrict__ query,
                                const float* __restrict__ mem,
                                const float* __restrict__ topVal,
                                const int*   __restrict__ topIdx,
                                float* __restrict__ out)      // [75*5]
{
    __shared__ float sP[N_WAY * D_DIM];
    __shared__ float sSum[N_WAY];
    __shared__ float sW[N_WAY];
    const int tid = threadIdx.x;

    if (tid < N_WAY) {
        float ws = 0.f;
#pragma unroll
        for (int k = 0; k < KTOP; ++k) ws += topVal[tid * KTOP + k];
        sW[tid] = ws;
        sSum[tid] = 0.f;
    }
    __syncthreads();

    for (int idx = tid; idx < N_WAY * D_DIM; idx += blockDim.x) {
        const int w = idx / D_DIM, dd = idx - w * D_DIM;
        float acc = 0.f;
#pragma unroll
        for (int k = 0; k < KTOP; ++k) {
            const float v = topVal[w * KTOP + k];
            const int i = topIdx[w * KTOP + k];
            const float* e = (i < N_SHOT)
                ? (support + ((size_t)i * N_WAY + w) * D_DIM)
                : (mem + (size_t)(i - N_SHOT) * D_DIM);
            acc = fmaf(v, e[dd], acc);
        }
        sP[idx] = acc / sW[w];
    }
    __syncthreads();
    for (int idx = tid; idx < N_WAY * D_DIM; idx += blockDim.x) {
        const int w = idx / D_DIM;
        const float v = sP[idx];
        atomicAdd(&sSum[w], v * v);
    }
    __syncthreads();
    for (int idx = tid; idx < N_WAY * D_DIM; idx += blockDim.x) {
        const int w = idx / D_DIM;
        sP[idx] /= fmaxf(sqrtf(sSum[w]), EPSF);
    }
    __syncthreads();

    for (int o = tid; o < N_QROWS * N_WAY; o += blockDim.x) {
        const int q = o / N_WAY, w = o - q * N_WAY;
        const float* qp = query + (size_t)q * D_DIM;
        float dot = 0.f;
        for (int dd = 0; dd < D_DIM; ++dd)
            dot = fmaf(qp[dd], sP[w * D_DIM + dd], dot);
        out[o] = dot * (1.0f / 64.0f);
    }
}

// ---------------------------------------------------------------------------
extern "C" void kernel_launch(void* const* d_in, const int* in_sizes, int n_in,
                              void* d_out, int out_size, void* d_ws, size_t ws_size,
                              hipStream_t stream) {
    const float* support = (const float*)d_in[0];  // [1,5,5,640]
    const float* query   = (const float*)d_in[1];  // [1,15,5,640]
    const float* memory  = (const float*)d_in[2];  // [M,640]
    float* out = (float*)d_out;

    const int M = in_sizes[2] / D_DIM;
    const int Mpad = ((M + 255) / 256) * 256;

    // workspace layout (256B-aligned slabs)
    char* base = (char*)d_ws;
    size_t off = 0;
    auto slab = [&](size_t bytes) {
        char* p = base + off;
        off = (off + bytes + 255) & ~(size_t)255;
        return p;
    };
    float* ws_proto  = (float*)slab(N_WAY * D_DIM * sizeof(float));
    float* ws_simsup = (float*)slab(N_WAY * N_SHOT * sizeof(float));
    float* ws_bpack  = (float*)slab(KSTEPS * 32 * 2 * sizeof(float));
    float* ws_topV   = (float*)slab(N_WAY * KTOP * sizeof(float));
    int*   ws_topI   = (int*)  slab(N_WAY * KTOP * sizeof(int));
    float* ws_sim    = (float*)slab((size_t)N_WAY * Mpad * sizeof(float));
    (void)ws_size; (void)n_in; (void)out_size;

    pn_prep_kernel<<<1, 256, 0, stream>>>(support, ws_proto, ws_simsup, ws_bpack);

    const int tiles = (M + 15) / 16;
    int blocks = (tiles + 15) / 16;        // 8 waves/block, ~2 tiles/wave
    if (blocks > 2048) blocks = 2048;
    if (blocks < 1) blocks = 1;
    pn_sim_kernel<<<blocks, 256, 0, stream>>>(memory, ws_bpack, ws_sim, M, Mpad);

    pn_topk_kernel<<<N_WAY, 256, 0, stream>>>(ws_sim, ws_simsup, ws_topV, ws_topI, M, Mpad);

    pn_final_kernel<<<1, 256, 0, stream>>>(support, query, memory, ws_topV, ws_topI, out);
}